// LightGCN_49160195670336
// MI455X (gfx1250) — compile-verified
//
#include <hip/hip_runtime.h>

#define D     64
#define HID   32
#define ATTR  8

typedef __attribute__((ext_vector_type(2))) float v2f;
typedef __attribute__((ext_vector_type(8))) float v8f;

// ---------------------------------------------------------------------------
// zero helper
// ---------------------------------------------------------------------------
__global__ void zero_kernel(float* __restrict__ p, long n) {
    long i = (long)blockIdx.x * blockDim.x + threadIdx.x;
    if (i < n) p[i] = 0.0f;
}

// ---------------------------------------------------------------------------
// x0 = [user_emb ; item_audio + artist_emb[artist_ids] + album_emb[album_ids]]
// one float2 per thread (rows are 256B aligned, c even -> 8B aligned)
// ---------------------------------------------------------------------------
__global__ void init_x_kernel(const float* __restrict__ user_emb,
                              const float* __restrict__ item_audio,
                              const float* __restrict__ artist_emb,
                              const float* __restrict__ album_emb,
                              const int*   __restrict__ artist_ids,
                              const int*   __restrict__ album_ids,
                              float* __restrict__ x,
                              long num_users, long n_nodes) {
    long idx = (long)blockIdx.x * blockDim.x + threadIdx.x;   // over n_nodes*32
    long row = idx >> 5;
    if (row >= n_nodes) return;
    int c = (int)(idx & 31) * 2;
    float2 v;
    if (row < num_users) {
        v = *(const float2*)(user_emb + row * D + c);
    } else {
        long i = row - num_users;
        float2 a  = *(const float2*)(item_audio + i * D + c);
        float2 ar = *(const float2*)(artist_emb + (long)artist_ids[i] * D + c);
        float2 al = *(const float2*)(album_emb  + (long)album_ids[i]  * D + c);
        v = make_float2(a.x + ar.x + al.x, a.y + ar.y + al.y);
    }
    *(float2*)(x + row * D + c) = v;
}

// ---------------------------------------------------------------------------
// edge_weight[e] = sigmoid( relu(ef[e] @ W1^T + b1) @ w2^T + b2 )
// weights staged in LDS once per block; one edge per thread (bandwidth bound)
// ---------------------------------------------------------------------------
__global__ void edge_mlp_kernel(const float* __restrict__ ef,
                                const float* __restrict__ w1,
                                const float* __restrict__ b1,
                                const float* __restrict__ w2,
                                const float* __restrict__ b2,
                                float* __restrict__ ew, long n_edges) {
    __shared__ float sW1[HID * ATTR];
    __shared__ float sB1[HID];
    __shared__ float sW2[HID];
    __shared__ float sB2;
    int t = threadIdx.x;
    if (t < HID * ATTR) sW1[t] = w1[t];
    if (t < HID) { sB1[t] = b1[t]; sW2[t] = w2[t]; }
    if (t == 0)  sB2 = b2[0];
    __syncthreads();

    long e = (long)blockIdx.x * blockDim.x + t;
    if (e >= n_edges) return;

    const float4* efv = (const float4*)(ef + e * ATTR);
    float4 f0 = efv[0], f1 = efv[1];
    float f[ATTR] = {f0.x, f0.y, f0.z, f0.w, f1.x, f1.y, f1.z, f1.w};

    float z = sB2;
#pragma unroll
    for (int h = 0; h < HID; ++h) {
        float acc = sB1[h];
#pragma unroll
        for (int k = 0; k < ATTR; ++k) acc = fmaf(sW1[h * ATTR + k], f[k], acc);
        z = fmaf(fmaxf(acc, 0.0f), sW2[h], z);
    }
    ew[e] = 1.0f / (1.0f + __expf(-z));
}

// ---------------------------------------------------------------------------
// deg[dst[e]] += ew[e]   (f32 global atomics, deg lives in L2)
// ---------------------------------------------------------------------------
__global__ void deg_accum_kernel(const float* __restrict__ ew,
                                 const int* __restrict__ dst,
                                 float* __restrict__ deg, long n_edges) {
    long e = (long)blockIdx.x * blockDim.x + threadIdx.x;
    if (e >= n_edges) return;
    atomicAdd(&deg[dst[e]], ew[e]);
}

// norm[e] = dinv[src]*w*dinv[dst]
__global__ void norm_kernel(const float* __restrict__ ew,
                            const int* __restrict__ src,
                            const int* __restrict__ dst,
                            const float* __restrict__ deg,
                            float* __restrict__ nrm, long n_edges) {
    long e = (long)blockIdx.x * blockDim.x + threadIdx.x;
    if (e >= n_edges) return;
    float w  = ew[e];
    float ds = deg[src[e]];
    float dd = deg[dst[e]];
    float a = ds > 0.0f ? rsqrtf(ds) : 0.0f;
    float b = dd > 0.0f ? rsqrtf(dd) : 0.0f;
    nrm[e] = a * w * b;
}

// ---------------------------------------------------------------------------
// one propagation layer: xout[dst] += xin[src] * norm   (segment_sum)
// one wave == one edge (64 cols as 32 float2 lanes); x stays L2-resident,
// the HBM-bound part is the src/dst/norm stream -> prefetch it ahead.
// ---------------------------------------------------------------------------
__global__ void propagate_kernel(const float* __restrict__ xin,
                                 float* __restrict__ xout,
                                 const int* __restrict__ src,
                                 const int* __restrict__ dst,
                                 const float* __restrict__ nrm, long n_edges) {
    long idx = (long)blockIdx.x * blockDim.x + threadIdx.x;   // n_edges*32
    long e = idx >> 5;
    if (e >= n_edges) return;
    int lane = (int)(idx & 31);
    if (lane == 0 && e + 2048 < n_edges) {
        __builtin_prefetch(&src[e + 2048], 0, 0);   // global_prefetch_b8
        __builtin_prefetch(&dst[e + 2048], 0, 0);
        __builtin_prefetch(&nrm[e + 2048], 0, 0);
    }
    float w = nrm[e];                 // wave-uniform
    if (w == 0.0f) return;            // uniform branch: whole wave skips
    int c = lane * 2;
    long so = (long)src[e] * D + c;
    long dofs = (long)dst[e] * D + c;
    float2 v = *(const float2*)(xin + so);
    atomicAdd(&xout[dofs + 0], v.x * w);
    atomicAdd(&xout[dofs + 1], v.y * w);
}

// ---------------------------------------------------------------------------
// Fused projection + alignment loss, full f32 via V_WMMA_F32_16X16X4_F32.
// One wave per 16x16 output tile of proj = item_audio @ W^T (K=64 -> 16 steps).
// A 16x4 layout: lane m=L&15, VGPRs span K (kh=L>>4): a[v] = A[m][kb+2kh+v].
// B  4x16 layout: lane n=L&15, b[v] = W[jb+n][kb+2kh+v]  (B[k][n] = W[n][k]).
// C/D: VGPR r, M = r + 8*kh, N = L&15.
// ---------------------------------------------------------------------------
__global__ void proj_loss_kernel(const float* __restrict__ item_audio,
                                 const float* __restrict__ Wproj,     // (64,64) row-major
                                 const float* __restrict__ item_final,
                                 float* __restrict__ loss_acc,
                                 long num_items) {
    long wave = ((long)blockIdx.x * blockDim.x + threadIdx.x) >> 5;
    int  lane = threadIdx.x & 31;
    long row_t = wave >> 2;          // 16-row tile index
    int  col_t = (int)(wave & 3);    // 16-col tile index (64/16)
    if (row_t * 16 >= num_items) return;          // wave-uniform

    long ib = row_t * 16;
    int  jb = col_t * 16;
    int  m  = lane & 15;
    int  kh = lane >> 4;

    const float* arow = item_audio + (ib + m) * D;
    const float* brow = Wproj + (long)(jb + m) * D;   // n = lane&15 == m

    v8f c = {};
#pragma unroll
    for (int kb = 0; kb < D; kb += 4) {
        int k0 = kb + 2 * kh;
        v2f a; a.x = arow[k0]; a.y = arow[k0 + 1];
        v2f b; b.x = brow[k0]; b.y = brow[k0 + 1];
        c = __builtin_amdgcn_wmma_f32_16x16x4_f32(
                /*neg_a=*/false, a, /*neg_b=*/false, b,
                /*c_mod=*/(short)0, c, /*reuse_a=*/false, /*reuse_b=*/false);
    }

    float s = 0.0f;
#pragma unroll
    for (int r = 0; r < 8; ++r) {
        int M = r + 8 * kh;
        float diff = item_final[(ib + M) * D + jb + m] - c[r];
        s += diff * diff;
    }
    // wave-level reduction
#pragma unroll
    for (int off = 16; off > 0; off >>= 1) s += __shfl_down(s, off, 32);
    if (lane == 0) atomicAdd(loss_acc, s);
}

__global__ void finalize_loss_kernel(const float* __restrict__ loss_acc,
                                     float* __restrict__ out, long denom) {
    if (blockIdx.x == 0 && threadIdx.x == 0)
        out[0] = loss_acc[0] / (float)denom;
}

// ---------------------------------------------------------------------------
// host-side orchestration
// ---------------------------------------------------------------------------
extern "C" void kernel_launch(void* const* d_in, const int* in_sizes, int n_in,
                              void* d_out, int out_size, void* d_ws, size_t ws_size,
                              hipStream_t stream) {
    const float* user_emb   = (const float*)d_in[0];
    const float* artist_emb = (const float*)d_in[1];
    const float* album_emb  = (const float*)d_in[2];
    const float* audio_proj = (const float*)d_in[3];
    const float* mlp_w1     = (const float*)d_in[4];
    const float* mlp_b1     = (const float*)d_in[5];
    const float* mlp_w2     = (const float*)d_in[6];
    const float* mlp_b2     = (const float*)d_in[7];
    const float* item_audio = (const float*)d_in[8];
    const float* edge_feat  = (const float*)d_in[9];
    const int*   artist_ids = (const int*)d_in[10];
    const int*   album_ids  = (const int*)d_in[11];
    const int*   edge_src   = (const int*)d_in[12];
    const int*   edge_dst   = (const int*)d_in[13];

    const long num_users = in_sizes[0] / D;
    const long num_items = in_sizes[8] / D;
    const long n_nodes   = num_users + num_items;
    const long n_edges   = in_sizes[12];

    float* ws   = (float*)d_ws;
    float* xa   = ws;                       // n_nodes * D
    float* xb   = xa + n_nodes * D;         // n_nodes * D
    float* ew   = xb + n_nodes * D;         // n_edges
    float* nrm  = ew + n_edges;             // n_edges
    float* deg  = nrm + n_edges;            // n_nodes
    float* loss = deg + n_nodes;            // 1

    float* out_nodes = (float*)d_out;       // n_nodes * D, then loss scalar

    const int B = 256;
    auto nblk = [](long n, int b) { return (unsigned)((n + b - 1) / b); };

    // x0 = [user_emb ; item_h]
    init_x_kernel<<<nblk(n_nodes * 32, B), B, 0, stream>>>(
        user_emb, item_audio, artist_emb, album_emb, artist_ids, album_ids,
        xa, num_users, n_nodes);

    // edge MLP -> edge_weight
    edge_mlp_kernel<<<nblk(n_edges, B), B, 0, stream>>>(
        edge_feat, mlp_w1, mlp_b1, mlp_w2, mlp_b2, ew, n_edges);

    // degree + normalization
    zero_kernel<<<nblk(n_nodes, B), B, 0, stream>>>(deg, n_nodes);
    deg_accum_kernel<<<nblk(n_edges, B), B, 0, stream>>>(ew, edge_dst, deg, n_edges);
    norm_kernel<<<nblk(n_edges, B), B, 0, stream>>>(ew, edge_src, edge_dst, deg, nrm, n_edges);

    // layer 1: xa -> xb
    zero_kernel<<<nblk(n_nodes * D, B), B, 0, stream>>>(xb, n_nodes * D);
    propagate_kernel<<<nblk(n_edges * 32, B), B, 0, stream>>>(xa, xb, edge_src, edge_dst, nrm, n_edges);
    // layer 2: xb -> xa
    zero_kernel<<<nblk(n_nodes * D, B), B, 0, stream>>>(xa, n_nodes * D);
    propagate_kernel<<<nblk(n_edges * 32, B), B, 0, stream>>>(xb, xa, edge_src, edge_dst, nrm, n_edges);
    // layer 3: xa -> d_out (final user/item features, contiguous in return order)
    zero_kernel<<<nblk(n_nodes * D, B), B, 0, stream>>>(out_nodes, n_nodes * D);
    propagate_kernel<<<nblk(n_edges * 32, B), B, 0, stream>>>(xa, out_nodes, edge_src, edge_dst, nrm, n_edges);

    // fused projection GEMM (WMMA f32) + MSE loss
    zero_kernel<<<1, 32, 0, stream>>>(loss, 1);
    long waves = ((num_items + 15) / 16) * 4;
    proj_loss_kernel<<<nblk(waves * 32, B), B, 0, stream>>>(
        item_audio, audio_proj, out_nodes + num_users * D, loss, num_items);
    finalize_loss_kernel<<<1, 1, 0, stream>>>(loss, (float*)d_out + n_nodes * D, num_items * D);
}